// MatConditionNumberLoss_64931315581116
// MI455X (gfx1250) — compile-verified
//
#include <hip/hip_runtime.h>
#include <hip/hip_bf16.h>
#include <math.h>

// Problem constants
constexpr int NM  = 128;              // matrix dim
constexpr int LDM = 132;              // padded LDS stride (132 % 64 == 4 -> conflict-free strided frag loads)
constexpr int NLOW = NM * (NM - 1) / 2;  // 8128 strictly-lower entries

typedef __attribute__((ext_vector_type(2))) float v2f;
typedef __attribute__((ext_vector_type(8))) float v8f;

// 128x128 f32 GEMM out of LDS using V_WMMA_F32_16X16X4_F32.
// Each of the 8 waves owns one 16-row block of C and all 8 column tiles.
// Results are staged in registers and written after a barrier, so C may
// alias A or Bm (in-place GEMM). TA/TB select transposed operand reads.
template <bool TA, bool TB>
__device__ __forceinline__ void mm128(const float* A, const float* Bm, float* C,
                                      int wave, int lane) {
  const int half = lane >> 4;   // 0: K pair {0,1}, 1: K pair {2,3} within a 4-wide K block
  const int l15  = lane & 15;
  const int ti   = wave * 16;   // this wave's row block

  v8f acc[8];
  v8f z = {};
#pragma unroll
  for (int t = 0; t < 8; ++t) acc[t] = z;

#pragma unroll 2
  for (int kb = 0; kb < 32; ++kb) {
    const int k0 = kb * 4 + half * 2;
    v2f a;
    if (TA) {   // A-fragment[m][k] = A[k][m]
      a.x = A[(k0    ) * LDM + ti + l15];
      a.y = A[(k0 + 1) * LDM + ti + l15];
    } else {
      a.x = A[(ti + l15) * LDM + k0    ];
      a.y = A[(ti + l15) * LDM + k0 + 1];
    }
#pragma unroll
    for (int t = 0; t < 8; ++t) {
      const int tj = t * 16;
      v2f b;
      if (TB) { // B-fragment[k][n] = Bm[n][k]
        b.x = Bm[(tj + l15) * LDM + k0    ];
        b.y = Bm[(tj + l15) * LDM + k0 + 1];
      } else {
        b.x = Bm[(k0    ) * LDM + tj + l15];
        b.y = Bm[(k0 + 1) * LDM + tj + l15];
      }
      // D = A(16x4) * B(4x16) + C  (f32)
      acc[t] = __builtin_amdgcn_wmma_f32_16x16x4_f32(
          false, a, false, b, (short)0, acc[t], false, false);
    }
  }

  __syncthreads();  // all source reads complete -> safe to overwrite (C may alias A/Bm)
#pragma unroll
  for (int t = 0; t < 8; ++t) {
    const int tj = t * 16;
#pragma unroll
    for (int r = 0; r < 8; ++r)  // C/D layout: VGPR r -> M = r (+8 for lanes 16..31)
      C[(ti + half * 8 + r) * LDM + tj + l15] = acc[t][r];
  }
  __syncthreads();
}

__global__ __launch_bounds__(256, 1)
void MatConditionNumberLoss_kernel(const float* __restrict__ DD,
                                   const float* __restrict__ Lent,
                                   const float* __restrict__ scale,
                                   float* __restrict__ out, float invB) {
  __shared__ float bufA[NM * LDM];   // L -> DD -> P
  __shared__ float bufB[NM * LDM];   // LL -> G = P^T P
  __shared__ float part[256];
  __shared__ float vvec[NM];
  __shared__ float wvec[NM];
  __shared__ float red[64];
  __shared__ float bc[1];

  const int tid  = threadIdx.x;
  const int wave = tid >> 5;
  const int lane = tid & 31;
  const int b    = blockIdx.x;

  const float* dd = DD + (size_t)b * (NM * NM);
  const float* le = Lent + (size_t)b * NLOW;

  // Warm the cache with this block's DD tile while we build L (global_prefetch_b8).
  __builtin_prefetch(dd + tid * 64, 0, 1);
  __builtin_prefetch(dd + tid * 64 + 32, 0, 1);

  // ---- Build L = scatter(L_entries)*scale + I into bufA ----
  const float s = scale[0];
  for (int i = tid; i < NM * LDM; i += 256) bufA[i] = 0.0f;
  __syncthreads();
  if (tid < NM) bufA[tid * LDM + tid] = 1.0f;
  for (int e = tid; e < NLOW; e += 256) {
    // invert e = i*(i-1)/2 + j  (strictly lower triangular, row-major order)
    int i = (int)((1.0f + sqrtf(1.0f + 8.0f * (float)e)) * 0.5f);
    while (i * (i - 1) / 2 > e) --i;
    while ((i + 1) * i / 2 <= e) ++i;
    const int j = e - i * (i - 1) / 2;
    bufA[i * LDM + j] = le[e] * s;
  }
  __syncthreads();

  // ---- LL = L * L^T  -> bufB ----
  mm128<false, true>(bufA, bufA, bufB, wave, lane);

  // ---- DD -> bufA (coalesced b128 loads) ----
  for (int f = tid; f < (NM * NM) / 4; f += 256) {
    const float4 q = ((const float4*)dd)[f];
    const int row = f >> 5;
    const int c4  = (f & 31) << 2;
    float* p = &bufA[row * LDM + c4];
    p[0] = q.x; p[1] = q.y; p[2] = q.z; p[3] = q.w;
  }
  __syncthreads();

  // ---- P = LL * DD -> bufA (in place over DD) ----
  mm128<false, false>(bufB, bufA, bufA, wave, lane);

  // ---- G = P^T * P -> bufB (in place over LL); sigma^2(P) = eig(G) ----
  mm128<true, false>(bufA, bufA, bufB, wave, lane);

  // ---- Power iteration: lambda_max(G) ----
  const int row = tid & 127;
  const int hh  = tid >> 7;  // two threads per row, 64-wide halves
  if (tid < NM) vvec[tid] = 1.0f;
  __syncthreads();

  float lmax = 0.0f;
  for (int it = 0; it < 24; ++it) {
    float acc = 0.0f;
    const float* g  = &bufB[row * LDM + hh * 64];
    const float* vv = &vvec[hh * 64];
#pragma unroll 8
    for (int j = 0; j < 64; ++j) acc += g[j] * vv[j];
    part[tid] = acc;
    __syncthreads();
    if (tid < NM) wvec[tid] = part[tid] + part[tid + 128];
    __syncthreads();
    if (tid < 64) red[tid] = wvec[tid] * wvec[tid] + wvec[tid + 64] * wvec[tid + 64];
    __syncthreads();
    if (tid < 32) {
      float x = red[tid] + red[tid + 32];
#pragma unroll
      for (int o = 16; o; o >>= 1) x += __shfl_xor(x, o, 32);
      if (tid == 0) bc[0] = x;
    }
    __syncthreads();
    const float nrm = sqrtf(bc[0]);
    lmax = nrm;
    const float rinv = (nrm > 0.0f) ? 1.0f / nrm : 0.0f;
    if (tid < NM) vvec[tid] = wvec[tid] * rinv;
    __syncthreads();
  }

  // ---- Shifted power iteration: lambda_min(G) = sh - lambda_max(sh*I - G) ----
  const float sh = lmax * 1.0002f + 1e-30f;
  if (tid < NM) vvec[tid] = (tid & 1) ? -1.0f : 1.0f;
  __syncthreads();

  float mu = 0.0f;
  for (int it = 0; it < 40; ++it) {
    float acc = 0.0f;
    const float* g  = &bufB[row * LDM + hh * 64];
    const float* vv = &vvec[hh * 64];
#pragma unroll 8
    for (int j = 0; j < 64; ++j) acc += g[j] * vv[j];
    part[tid] = acc;
    __syncthreads();
    if (tid < NM) wvec[tid] = sh * vvec[tid] - (part[tid] + part[tid + 128]);
    __syncthreads();
    if (tid < 64) red[tid] = wvec[tid] * wvec[tid] + wvec[tid + 64] * wvec[tid + 64];
    __syncthreads();
    if (tid < 32) {
      float x = red[tid] + red[tid + 32];
#pragma unroll
      for (int o = 16; o; o >>= 1) x += __shfl_xor(x, o, 32);
      if (tid == 0) bc[0] = x;
    }
    __syncthreads();
    const float nrm = sqrtf(bc[0]);
    mu = nrm;
    const float rinv = (nrm > 0.0f) ? 1.0f / nrm : 0.0f;
    if (tid < NM) vvec[tid] = wvec[tid] * rinv;
    __syncthreads();
  }

  if (tid == 0) {
    const float lmin = fmaxf(sh - mu, 1e-20f);
    const float cond = sqrtf(lmax / lmin);  // sigma_max/sigma_min
    atomicAdd(out, cond * invB);
  }
}

__global__ void zero_out_kernel(float* o) {
  if (threadIdx.x == 0) o[0] = 0.0f;
}

extern "C" void kernel_launch(void* const* d_in, const int* in_sizes, int n_in,
                              void* d_out, int out_size, void* d_ws, size_t ws_size,
                              hipStream_t stream) {
  const float* DD    = (const float*)d_in[0];
  const float* Lent  = (const float*)d_in[1];
  const float* scale = (const float*)d_in[2];
  float* out = (float*)d_out;

  const int B = in_sizes[0] / (NM * NM);  // 2048
  zero_out_kernel<<<1, 32, 0, stream>>>(out);
  MatConditionNumberLoss_kernel<<<B, 256, 0, stream>>>(DD, Lent, scale, out,
                                                       1.0f / (float)B);
}